// AdaptiveWaveletTransform_81630148428272
// MI455X (gfx1250) — compile-verified
//
#include <hip/hip_runtime.h>

// CDNA5 / gfx1250, wave32. D = A(16x4) x B(4x16) + C via V_WMMA_F32_16X16X4_F32.
typedef __attribute__((ext_vector_type(2))) float v2f;
typedef __attribute__((ext_vector_type(4))) float v4f;
typedef __attribute__((ext_vector_type(8))) float v8f;

#define NBATCH 32
#define SEQLEN 4096
#define NFEAT  64

// Banded-filter fragment ROM. For lane (mn, khalf) and chunk c, the A fragments are
//   aLo = {H0[2e], H0[2e+1]}, aHi = {H1[2e], H1[2e+1]},  e = 2c + khalf - mn  (zero outside [0,4))
// Row (e+15) holds {H0[2e], H0[2e+1], H1[2e], H1[2e+1]}; rows 0..14 and 19..35 are zero.
// Accessed as base + 32B*c immediate-offset b128 loads. H0/H1 = time-reversed db4 dec_lo/dec_hi.
__device__ __constant__ __attribute__((aligned(16))) float kBand[36][4] = {
  {0,0,0,0},{0,0,0,0},{0,0,0,0},{0,0,0,0},{0,0,0,0},
  {0,0,0,0},{0,0,0,0},{0,0,0,0},{0,0,0,0},{0,0,0,0},
  {0,0,0,0},{0,0,0,0},{0,0,0,0},{0,0,0,0},{0,0,0,0},
  { 0.23037781330885523f,  0.7148465705525415f,  -0.010597401784997278f, -0.032883011666982945f},
  { 0.6308807679295904f,  -0.02798376941698385f,  0.030841381835986965f,  0.18703481171888114f },
  {-0.18703481171888114f,  0.030841381835986965f,-0.02798376941698385f,  -0.6308807679295904f  },
  { 0.032883011666982945f,-0.010597401784997278f, 0.7148465705525415f,   -0.23037781330885523f }
  // rows 19..35 zero-initialized
};

// One wave32 computes a 16(position) x 16(feature) tile of both lo and hi coefficients.
//   out[j] = sum_{t=0..7} H[t] * x[2j - 3 + t]   (left zero-pad == 3 at every level)
// Banded decomposition over 10 chunks of K=4:
//   A_c[m][k] = h[4c + k - 2m],  B_c[k][n] = in[2*jbase - 3 + 4c + k][f0+n]  (zero OOB)
__global__ void dwt_level_wmma(const float* __restrict__ in,
                               float* __restrict__ lo_out,
                               float* __restrict__ hi_out,
                               int Tin, int Tout, int ntiles)
{
  const int lane  = threadIdx.x & 31;   // lane within wave32
  const int wave  = threadIdx.x >> 5;   // 8 waves per block -> 8 consecutive j-tiles
  const int mn    = lane & 15;          // A-row m  /  B- & D-column n
  const int khalf = lane >> 4;          // 0 -> K {0,1}, 1 -> K {2,3}

  const int jtile = blockIdx.x * 8 + wave;
  if (jtile >= ntiles) return;          // wave-uniform exit; active waves keep EXEC all-ones
  const int jbase = jtile * 16;
  const int b     = blockIdx.z;
  const int f     = blockIdx.y * 16 + mn;

  // A fragments: one base address, 10 immediate-offset b128 loads from the 576-B ROM.
  v2f aLo[10], aHi[10];
  {
    const v4f* bp = (const v4f*)kBand + (khalf - mn + 15);  // rows advance by 2 per chunk
#pragma unroll
    for (int c = 0; c < 10; ++c) {
      v4f t = bp[2 * c];
      aLo[c] = __builtin_shufflevector(t, t, 0, 1);         // sub-register pair
      aHi[c] = __builtin_shufflevector(t, t, 2, 3);
    }
  }

  // B fragments: rows rlo + 4c + 2*khalf (+1 for .y), feature column f.
  const int  rlo     = 2 * jbase - 3;            // first input row touched (chunk 0, k=0)
  const long rowBase = (long)b * Tin;
  v2f bf[10];

  if (rlo >= 0 && rlo + 39 < Tin) {
    // Interior tile (wave-uniform): one address, 20 immediate-offset loads that clause.
    const float* p = in + ((rowBase + rlo + 2 * khalf) << 6) + f;
#pragma unroll
    for (int c = 0; c < 10; ++c) {
      bf[c].x = p[c * 256];                      // +4 rows per chunk  (1024 B step)
      bf[c].y = p[c * 256 + 64];                 // +1 row             (+256 B)
    }
  } else {
    // Boundary tile: clamp addresses (always safe), zero OOB values via select. No EXEC games.
#pragma unroll
    for (int c = 0; c < 10; ++c) {
      const int r0  = rlo + 4 * c + 2 * khalf;
      const int r1  = r0 + 1;
      const int r0c = min(max(r0, 0), Tin - 1);
      const int r1c = min(max(r1, 0), Tin - 1);
      const float x0 = in[((rowBase + r0c) << 6) + f];
      const float x1 = in[((rowBase + r1c) << 6) + f];
      bf[c].x = (r0 == r0c) ? x0 : 0.0f;
      bf[c].y = (r1 == r1c) ? x1 : 0.0f;
    }
  }

  // Same B fragment feeds both accumulators (lo via H0-band, hi via H1-band).
  v8f dLo = {};
  v8f dHi = {};
#pragma unroll
  for (int c = 0; c < 10; ++c) {
    dLo = __builtin_amdgcn_wmma_f32_16x16x4_f32(false, aLo[c], false, bf[c], (short)0,
                                                dLo, false, false);
    dHi = __builtin_amdgcn_wmma_f32_16x16x4_f32(false, aHi[c], false, bf[c], (short)0,
                                                dHi, false, false);
  }

  // D layout: VGPR v -> M = v + 8*khalf, N = mn. Intermediates stay RT (L2-resident for next level).
  const long obase = (long)b * Tout;
  float* plo = lo_out + ((obase + jbase + 8 * khalf) << 6) + f;
  float* phi = hi_out + ((obase + jbase + 8 * khalf) << 6) + f;
  if (jbase + 16 <= Tout) {
    // Full tile (wave-uniform): immediate-offset stores.
#pragma unroll
    for (int v = 0; v < 8; ++v) {
      plo[v * 64] = dLo[v];
      phi[v * 64] = dHi[v];
    }
  } else {
#pragma unroll
    for (int v = 0; v < 8; ++v) {
      if (jbase + v + 8 * khalf < Tout) {
        plo[v * 64] = dLo[v];
        phi[v * 64] = dHi[v];
      }
    }
  }
}

// Compact-coefficient workspace layout (floats); plane size = 32*64*len = 2048*len.
__device__ __constant__ int  kLen[5]   = {2051, 1029, 518, 262, 134};
__device__ __constant__ long kOffLo[5] = {0,        8400896, 12615680, 14737408, 15810560};
__device__ __constant__ long kOffHi[5] = {4200448, 10508288, 13676544, 15273984, 16084992};

// Writes every element of d_out: [approx | details(5) | high_freq | low_freq], each (B,S,F).
// Output is written once and never re-read -> non-temporal stores keep ws/input in L2.
__global__ void awt_assemble(const float* __restrict__ ws,
                             const float* __restrict__ scores,
                             float* __restrict__ out)
{
  const long N = (long)NBATCH * SEQLEN * NFEAT;
  long idx = (long)blockIdx.x * blockDim.x + threadIdx.x;
  if (idx >= N) return;
  const int f = (int)(idx & 63);
  const int s = (int)((idx >> 6) & (SEQLEN - 1));
  const int b = (int)(idx >> 18);

  // level = clip(2 + round_half_even(score * 3), 2, 5)
  int lvl = 2 + (int)rintf(scores[f] * 3.0f);
  lvl = lvl < 2 ? 2 : (lvl > 5 ? 5 : lvl);

  float approx = 0.0f;
  {
    const int L = kLen[lvl - 1];
    if (s < L) approx = ws[kOffLo[lvl - 1] + (((long)b * L + s) << 6) + f];
  }

  float hsum = 0.0f;
#pragma unroll
  for (int l = 0; l < 5; ++l) {
    float d = 0.0f;
    const int L = kLen[l];
    if (l < lvl && s < L) d = ws[kOffHi[l] + (((long)b * L + s) << 6) + f];
    __builtin_nontemporal_store(d, &out[N * (1 + l) + idx]);   // details plane l
    hsum += d;
  }
  __builtin_nontemporal_store(approx, &out[idx]);              // approx
  __builtin_nontemporal_store(hsum,   &out[6 * N + idx]);      // high_freq
  __builtin_nontemporal_store(approx, &out[7 * N + idx]);      // low_freq
}

extern "C" void kernel_launch(void* const* d_in, const int* in_sizes, int n_in,
                              void* d_out, int out_size, void* d_ws, size_t ws_size,
                              hipStream_t stream) {
  (void)in_sizes; (void)n_in; (void)out_size; (void)ws_size;
  const float* x      = (const float*)d_in[0];   // (32, 4096, 64) f32
  const float* scores = (const float*)d_in[1];   // (64,) f32
  float* out = (float*)d_out;
  float* ws  = (float*)d_ws;                     // needs 16,359,424 floats (~65.4 MB)

  static const int  lens[5]  = {2051, 1029, 518, 262, 134};
  static const long offLo[5] = {0,        8400896, 12615680, 14737408, 15810560};
  static const long offHi[5] = {4200448, 10508288, 13676544, 15273984, 16084992};

  // 5-level cascade: level l reads lo_{l-1} (level 1 reads x), writes compact lo_l / hi_l.
  const float* cur = x;
  int Tin = SEQLEN;
  for (int l = 0; l < 5; ++l) {
    const int Tout   = lens[l];
    const int ntiles = (Tout + 15) / 16;
    dim3 grid((ntiles + 7) / 8, NFEAT / 16, NBATCH);  // 8 waves/block, 4 feature tiles, 32 batches
    dwt_level_wmma<<<grid, 256, 0, stream>>>(cur, ws + offLo[l], ws + offHi[l],
                                             Tin, Tout, ntiles);
    cur = ws + offLo[l];
    Tin = Tout;
  }

  // Assemble all 8 (B,S,F) output planes (256 MB of writes dominates; ~18 us at 23.3 TB/s).
  const long N = (long)NBATCH * SEQLEN * NFEAT;
  const int  threads = 256;
  const long blocks  = (N + threads - 1) / threads;
  awt_assemble<<<dim3((unsigned)blocks), threads, 0, stream>>>(ws, scores, out);
}